// KeypointLoss_67259187855796
// MI455X (gfx1250) — compile-verified
//
#include <hip/hip_runtime.h>
#include <cstdint>

// ---------------------------------------------------------------------------
// KeypointLoss on MI455X (gfx1250)
//   loss = mean((pred_hm - gauss_target)^2) + 0.5 * BCE(pred_vis, tgt_vis)
// Memory-bound: one 160.5MB stream of pred_heatmaps -> ~6.9us floor @23.3TB/s.
// Gaussian target is separable: tgt(y,x) = gy[y]*gx[x]  (418K exps vs 40.1M).
// Stage 1: 1 block per (b,k) slice, async global->LDS 4-deep pipeline,
//          per-wave s_wait_asynccnt (no block barriers in the hot loop),
//          block partial-sum reduced on the matrix pipe (WMMA f32 16x16x4,
//          B = ones => exact f32 row-sum accumulate).
// Stage 2: fold 1088 partials + BCE into the scalar.
// ---------------------------------------------------------------------------

namespace {

constexpr int   Bn = 64, Kn = 17, Hn = 192, Wn = 192;
constexpr int   SLICE   = Hn * Wn;        // 36864 floats per (b,k)
constexpr int   NBK     = Bn * Kn;        // 1088 slices
constexpr int   THREADS = 256;            // 8 waves (wave32)
constexpr int   CHUNK   = THREADS * 4;    // 1024 floats = 4KB per chunk
constexpr int   NCH     = SLICE / CHUNK;  // 36 chunks per slice
constexpr int   PIPE    = 4;              // async pipeline depth (per wave)
constexpr float INV_2SIG2 = 0.125f;       // 1/(2*sigma^2), sigma=2

typedef __attribute__((ext_vector_type(2))) float v2f;
typedef __attribute__((ext_vector_type(8))) float v8f;

// Flat shared-aperture address: ISA 10.2 — LDS aperture lives in addr[63:32],
// addr[31:0] is the logical LDS byte offset the async-load VDST operand wants.
__device__ __forceinline__ uint32_t lds_off_of(const void* p) {
  return (uint32_t)(uintptr_t)p;
}

// CDNA5 async copy global -> LDS (tracked by ASYNCcnt). Inline asm so it is
// portable across ROCm7.2 / upstream clang regardless of builtin arity.
__device__ __forceinline__ void async_load_b128_to_lds(uint32_t lds, const void* g) {
  asm volatile("global_load_async_to_lds_b128 %0, %1, off"
               :: "v"(lds), "v"((uint64_t)(uintptr_t)g)
               : "memory");
}

__device__ __forceinline__ void wait_async_le(int n) {
  switch (n) {
    case 0:  asm volatile("s_wait_asynccnt 0x0" ::: "memory"); break;
    case 1:  asm volatile("s_wait_asynccnt 0x1" ::: "memory"); break;
    case 2:  asm volatile("s_wait_asynccnt 0x2" ::: "memory"); break;
    default: asm volatile("s_wait_asynccnt 0x3" ::: "memory"); break;
  }
}

__global__ __launch_bounds__(THREADS) void
stage1_heatmap_sq(const float* __restrict__ pred,
                  const float* __restrict__ kpts,
                  float* __restrict__ partial) {
  __shared__ __align__(16) float gx[Wn];
  __shared__ float gy[Hn];
  __shared__ __align__(16) float tile[PIPE][CHUNK];
  __shared__ float red[THREADS];

  const int tid = threadIdx.x;
  const int bk  = blockIdx.x;             // one (b,k) slice per block

  // Per-slice Gaussian parameters (uniform across the block).
  const float kx = kpts[bk * 3 + 0];
  const float ky = kpts[bk * 3 + 1];
  const float kv = kpts[bk * 3 + 2];
  const int  hx = (int)floorf(kx * (float)Wn);
  const int  hy = (int)floorf(ky * (float)Hn);
  const bool valid = (kv > 0.0f) && (hx >= 0) && (hx < Wn) && (hy >= 0) && (hy < Hn);

  // Separable Gaussian: tgt(y,x) = gy[y] * gx[x]; invalid -> tgt == 0.
  for (int i = tid; i < Wn; i += THREADS) {
    const float dx = (float)(i - hx);
    const float dy = (float)(i - hy);
    gx[i] = valid ? __expf(-dx * dx * INV_2SIG2) : 0.0f;
    gy[i] = __expf(-dy * dy * INV_2SIG2);
  }
  __syncthreads();

  const float* src = pred + (size_t)bk * SLICE;

  // Prologue: prime PIPE-1 chunks.
  for (int c = 0; c < PIPE - 1; ++c)
    async_load_b128_to_lds(lds_off_of(&tile[c][tid * 4]),
                           src + c * CHUNK + tid * 4);

  // Steady state: each wave consumes only LDS its own lanes wrote, ASYNCcnt is
  // per-wave and async loads retire in order, so no __syncthreads needed here.
  float acc = 0.0f;
  for (int c = 0; c < NCH; ++c) {
    const int ahead = c + PIPE - 1;
    if (ahead < NCH)
      async_load_b128_to_lds(lds_off_of(&tile[ahead & (PIPE - 1)][tid * 4]),
                             src + ahead * CHUNK + tid * 4);
    const int rem = NCH - 1 - c;          // chunks newer than c still possibly in flight
    wait_async_le(rem < PIPE - 1 ? rem : PIPE - 1);   // oldest (c) now complete

    const int flat = c * CHUNK + tid * 4; // 192 % 4 == 0 -> quad stays in one row
    const int y = flat / Wn;
    const int x = flat % Wn;
    const float  gyv = gy[y];
    const float4 p = *(const float4*)&tile[c & (PIPE - 1)][tid * 4];
    const float4 g = *(const float4*)&gx[x];
    const float e0 = p.x - gyv * g.x;
    const float e1 = p.y - gyv * g.y;
    const float e2 = p.z - gyv * g.z;
    const float e3 = p.w - gyv * g.w;
    acc += e0 * e0; acc += e1 * e1; acc += e2 * e2; acc += e3 * e3;
  }

  red[tid] = acc;
  __syncthreads();

#if __has_builtin(__builtin_amdgcn_wmma_f32_16x16x4_f32)
  // Block reduction on the matrix pipe: D = A(16x4 f32) x ones(4x16) + C gives
  // exact f32 row-sums; 4 WMMAs fold all 256 partials into one 16x16 C tile
  // whose columns are identical. Wave 0 only -> EXEC all 1s as WMMA requires.
  if (tid < 32) {
    v8f cacc = {};
    v2f ones; ones[0] = 1.0f; ones[1] = 1.0f;
#pragma unroll
    for (int j = 0; j < THREADS / 64; ++j) {
      v2f a;
      a[0] = red[j * 64 + tid];
      a[1] = red[j * 64 + 32 + tid];
      cacc = __builtin_amdgcn_wmma_f32_16x16x4_f32(
          /*neg_a=*/false, a, /*neg_b=*/false, ones,
          /*c_mod=*/(short)0, cacc, /*reuse_a=*/false, /*reuse_b=*/false);
    }
    // C layout: lane n holds column n for M=0..7 (lanes 0-15) / M=8..15 (16-31).
    float s = cacc[0] + cacc[1] + cacc[2] + cacc[3] +
              cacc[4] + cacc[5] + cacc[6] + cacc[7];
    s += __shfl_down(s, 16);              // wave32: fold upper half-wave
    if (tid == 0) partial[bk] = s;
  }
#else
  for (int off = THREADS / 2; off > 0; off >>= 1) {
    if (tid < off) red[tid] += red[tid + off];
    __syncthreads();
  }
  if (tid == 0) partial[bk] = red[0];
#endif
}

__global__ __launch_bounds__(THREADS) void
stage2_finalize(const float* __restrict__ partial,
                const float* __restrict__ pv,
                const float* __restrict__ tv,
                float* __restrict__ out) {
  __shared__ float r1[THREADS];
  __shared__ float r2[THREADS];
  const int tid = threadIdx.x;
  float s1 = 0.0f, s2 = 0.0f;
  for (int i = tid; i < NBK; i += THREADS) {
    s1 += partial[i];
    const float p = pv[i];
    const float t = tv[i];
    s2 += t * __logf(p) + (1.0f - t) * __logf(1.0f - p);
  }
  r1[tid] = s1; r2[tid] = s2;
  __syncthreads();
  for (int off = THREADS / 2; off > 0; off >>= 1) {
    if (tid < off) { r1[tid] += r1[tid + off]; r2[tid] += r2[tid + off]; }
    __syncthreads();
  }
  if (tid == 0) {
    const float hm_mean = r1[0] * (1.0f / ((float)NBK * (float)SLICE));
    const float bce     = -r2[0] * (1.0f / (float)NBK);
    out[0] = hm_mean + 0.5f * bce;
  }
}

} // namespace

extern "C" void kernel_launch(void* const* d_in, const int* in_sizes, int n_in,
                              void* d_out, int out_size, void* d_ws, size_t ws_size,
                              hipStream_t stream) {
  (void)in_sizes; (void)n_in; (void)out_size; (void)ws_size;
  const float* pred = (const float*)d_in[0];   // [64,17,192,192] f32
  const float* pv   = (const float*)d_in[1];   // [64,17]         f32
  const float* kpts = (const float*)d_in[2];   // [64,17,3]       f32
  const float* tv   = (const float*)d_in[3];   // [64,17]         f32
  float* out     = (float*)d_out;              // scalar f32
  float* partial = (float*)d_ws;               // 1088 block partials

  stage1_heatmap_sq<<<NBK, THREADS, 0, stream>>>(pred, kpts, partial);
  stage2_finalize<<<1, THREADS, 0, stream>>>(partial, pv, tv, out);
}